// HRLModel_34875134443709
// MI455X (gfx1250) — compile-verified
//
#include <hip/hip_runtime.h>
#include <math.h>

typedef __attribute__((ext_vector_type(2))) float v2f;
typedef __attribute__((ext_vector_type(8))) float v8f;

#define Bn      32
#define Dn      128
#define DS      129          // padded LDS row stride (bank-conflict free: 129%64==1)
#define NPAIR   (Bn*Bn)      // 1024
#define NSTEP   (Bn-1)       // 31
#define THREADS 256

__device__ __forceinline__ float sigmoidf_(float x) { return 1.0f / (1.0f + __expf(-x)); }

__global__ __launch_bounds__(THREADS, 1)
void treelstm_parse_kernel(const float* __restrict__ x_emb,      // (48,128)
                           const float* __restrict__ W_leaf,     // (128,256)
                           const float* __restrict__ b_leaf,     // (256)
                           const float* __restrict__ W_tree,     // (256,640)
                           const float* __restrict__ b_tree,     // (640)
                           const float* __restrict__ W_a1,       // (256,128)
                           const float* __restrict__ b_a1,       // (128)
                           const float* __restrict__ W_a2,       // (128,1)
                           const float* __restrict__ b_a2,       // (1)
                           const int*   __restrict__ bottom_idx, // (32)
                           float* __restrict__ out)               // 4160 floats
{
    __shared__ float sh_hidden[Bn * DS];
    __shared__ float sh_cell  [Bn * DS];
    __shared__ float sh_S1    [Bn * DS];
    __shared__ float sh_S2    [Bn * DS];
    __shared__ float sh_score [NPAIR];
    __shared__ float sh_g     [5 * Dn];
    __shared__ float sh_mask  [Bn];
    __shared__ float sh_redv  [THREADS];
    __shared__ int   sh_redi  [THREADS];
    __shared__ float sh_rA    [THREADS];
    __shared__ float sh_rB    [THREADS];
    __shared__ float sh_rC    [THREADS];
    __shared__ float sh_lps   [NSTEP];
    __shared__ float sh_ents  [NSTEP];
    __shared__ int   sh_heads [NSTEP];
    __shared__ int   sh_deps  [NSTEP];
    __shared__ int   sh_selh, sh_seld;

    const int t    = threadIdx.x;
    const int lane = t & 31;
    const int wave = t >> 5;
    const int half = lane >> 4;   // 0: lanes 0-15, 1: lanes 16-31
    const int l16  = lane & 15;

    // ---------------- Leaf projection: hc = x[bottom_idx] @ W_leaf + b_leaf ----
    // thread t owns output column c = t (0..255) for every batch row b.
    {
        const float bl = b_leaf[t];
        for (int b = 0; b < Bn; ++b) {
            const int row = bottom_idx[b];
            float acc = bl;
            for (int d = 0; d < Dn; ++d)
                acc += x_emb[row * Dn + d] * W_leaf[d * (2 * Dn) + t];
            if (t < Dn) sh_hidden[b * DS + t]        = acc;
            else        sh_cell  [b * DS + (t - Dn)] = acc;
        }
        if (t < Bn) sh_mask[t] = 1.0f;
    }
    __syncthreads();

    const float ba2 = b_a2[0];

    // ---------------- 31 sequential merge steps --------------------------------
    for (int step = 0; step < NSTEP; ++step) {
        // (a) S1 = hidden @ W_a1[0:128,:], S2 = hidden @ W_a1[128:256,:]
        //     via V_WMMA_F32_16X16X4_F32. 32 output tiles (2 mats x 2 Mtiles x 8 Ntiles),
        //     8 wave32s -> 4 tiles per wave. EXEC all-ones (no divergence here).
        for (int q = 0; q < 4; ++q) {
            const int T    = wave * 4 + q;
            const int smat = T >> 4;          // 0 -> S1, 1 -> S2
            const int rem  = T & 15;
            const int m0   = (rem >> 3) * 16; // 0 or 16
            const int n0   = (rem & 7) * 16;  // 0..112
            const float* __restrict__ Wb = W_a1 + smat * Dn * Dn;

            v8f acc = {};
#pragma unroll
            for (int k = 0; k < Dn; k += 4) {
                v2f av, bv;
                // A 16x4 f32 fragment: lane l16 row m0+l16; half selects K pair {0,1} vs {2,3}
                av.x = sh_hidden[(m0 + l16) * DS + k + half * 2 + 0];
                av.y = sh_hidden[(m0 + l16) * DS + k + half * 2 + 1];
                // B 4x16 f32 fragment: row K across lanes (VGPR0: K=0/2, VGPR1: K=1/3)
                bv.x = Wb[(k + half * 2 + 0) * Dn + n0 + l16];
                bv.y = Wb[(k + half * 2 + 1) * Dn + n0 + l16];
                acc = __builtin_amdgcn_wmma_f32_16x16x4_f32(
                        false, av, false, bv, (short)0, acc, false, false);
            }
            float* __restrict__ Sd = smat ? sh_S2 : sh_S1;
#pragma unroll
            for (int r = 0; r < 8; ++r)   // C/D layout: VGPR r -> M = m0 + r + half*8
                Sd[(m0 + half * 8 + r) * DS + n0 + l16] = acc[r];
        }
        __syncthreads();

        // (b) pairwise arc scores: score[i,j] = relu(S1[i]+S2[j]+b_a1) . W_a2 + b_a2
        for (int r = 0; r < 4; ++r) {
            const int p = t + r * THREADS;      // flat pair index
            const int i = p >> 5, j = p & 31;
            float acc = ba2;
            for (int k = 0; k < Dn; ++k) {
                const float v = sh_S1[i * DS + k] + sh_S2[j * DS + k] + b_a1[k];
                acc += fmaxf(v, 0.0f) * W_a2[k];
            }
            sh_score[p] = acc;
        }
        __syncthreads();

        // (c) masked argmax (first-index tie-break, as jnp.argmax)
        float best = -INFINITY; int bidx = NPAIR;
        for (int r = 0; r < 4; ++r) {
            const int p = t + r * THREADS;
            const int i = p >> 5, j = p & 31;
            if (i != j && sh_mask[j] > 0.0f) {
                const float s = sh_score[p];
                if (s > best || (s == best && p < bidx)) { best = s; bidx = p; }
            }
        }
        sh_redv[t] = best; sh_redi[t] = bidx;
        __syncthreads();
        for (int s = THREADS / 2; s > 0; s >>= 1) {
            if (t < s) {
                const float ov = sh_redv[t + s]; const int oi = sh_redi[t + s];
                if (ov > sh_redv[t] || (ov == sh_redv[t] && oi < sh_redi[t])) {
                    sh_redv[t] = ov; sh_redi[t] = oi;
                }
            }
            __syncthreads();
        }
        const float m   = sh_redv[0];
        const int   idx = sh_redi[0];

        // masked log-softmax stats: Z = sum e^(s-m), T = sum e^(s-m)*(s-m), C = n_valid
        float Zp = 0.0f, Tp = 0.0f, Cp = 0.0f;
        for (int r = 0; r < 4; ++r) {
            const int p = t + r * THREADS;
            const int i = p >> 5, j = p & 31;
            if (i != j && sh_mask[j] > 0.0f) {
                const float s = sh_score[p];
                const float e = __expf(s - m);
                Zp += e; Tp += e * (s - m); Cp += 1.0f;
            }
        }
        sh_rA[t] = Zp; sh_rB[t] = Tp; sh_rC[t] = Cp;
        __syncthreads();
        for (int s = THREADS / 2; s > 0; s >>= 1) {
            if (t < s) {
                sh_rA[t] += sh_rA[t + s];
                sh_rB[t] += sh_rB[t + s];
                sh_rC[t] += sh_rC[t + s];
            }
            __syncthreads();
        }
        if (t == 0) {
            const float Z = sh_rA[0], Tsum = sh_rB[0], Cn = sh_rC[0];
            const float logZ = __logf(Z);
            sh_lps[step]  = sh_score[idx] - m - logZ;             // == -logZ
            sh_ents[step] = -(Tsum / Z - logZ) / __logf(Cn);
            const int h = idx >> 5, dd = idx & 31;
            sh_heads[step] = h; sh_deps[step] = dd;
            sh_selh = h; sh_seld = dd;
            sh_mask[dd] = 0.0f;
        }
        __syncthreads();
        const int h  = sh_selh;
        const int dd = sh_seld;

        // (d) gates for ONLY the selected pair: g = h_h @ Wt_top + h_d @ Wt_bot + b_tree
        {
            float a0 = b_tree[t];
            float a1 = b_tree[t + 256];
            float a2 = (t < Dn) ? b_tree[t + 512] : 0.0f;
            for (int k = 0; k < 2 * Dn; ++k) {
                const float hv = (k < Dn) ? sh_hidden[h * DS + k]
                                          : sh_hidden[dd * DS + (k - Dn)];
                const float* __restrict__ wr = W_tree + k * (5 * Dn);
                a0 += hv * wr[t];
                a1 += hv * wr[t + 256];
                if (t < Dn) a2 += hv * wr[t + 512];
            }
            sh_g[t] = a0; sh_g[t + 256] = a1;
            if (t < Dn) sh_g[t + 512] = a2;
        }
        __syncthreads();

        // TreeLSTM cell update of row h; mask row dd already cleared
        if (t < Dn) {
            const float gi  = sh_g[0 * Dn + t];
            const float gfl = sh_g[1 * Dn + t];
            const float gfr = sh_g[2 * Dn + t];
            const float gg  = sh_g[3 * Dn + t];
            const float go  = sh_g[4 * Dn + t];
            const float cn  = sigmoidf_(gi)  * tanhf(gg)
                            + sigmoidf_(gfl) * sh_cell[h  * DS + t]
                            + sigmoidf_(gfr) * sh_cell[dd * DS + t];
            const float hn  = sigmoidf_(go) * tanhf(cn);
            sh_cell  [h * DS + t] = cn;
            sh_hidden[h * DS + t] = hn;
        }
        __syncthreads();
    }

    // ---------------- outputs: hidden | sum(lps) | mean(ents) | edges ----------
    for (int o = t; o < Bn * Dn; o += THREADS)
        out[o] = sh_hidden[(o >> 7) * DS + (o & 127)];
    if (t == 0) {
        float lpsum = 0.0f, entsum = 0.0f;
        for (int s = 0; s < NSTEP; ++s) { lpsum += sh_lps[s]; entsum += sh_ents[s]; }
        out[Bn * Dn + 0] = lpsum;
        out[Bn * Dn + 1] = entsum / (float)NSTEP;
    }
    if (t < NSTEP) {
        out[Bn * Dn + 2 + 2 * t + 0] = (float)bottom_idx[sh_heads[t]];
        out[Bn * Dn + 2 + 2 * t + 1] = (float)bottom_idx[sh_deps[t]];
    }
}

extern "C" void kernel_launch(void* const* d_in, const int* in_sizes, int n_in,
                              void* d_out, int out_size, void* d_ws, size_t ws_size,
                              hipStream_t stream) {
    const float* x_emb      = (const float*)d_in[0];
    const float* W_leaf     = (const float*)d_in[1];
    const float* b_leaf     = (const float*)d_in[2];
    const float* W_tree     = (const float*)d_in[3];
    const float* b_tree     = (const float*)d_in[4];
    const float* W_a1       = (const float*)d_in[5];
    const float* b_a1       = (const float*)d_in[6];
    const float* W_a2       = (const float*)d_in[7];
    const float* b_a2       = (const float*)d_in[8];
    const int*   bottom_idx = (const int*)d_in[9];
    float* out = (float*)d_out;

    treelstm_parse_kernel<<<dim3(1), dim3(THREADS), 0, stream>>>(
        x_emb, W_leaf, b_leaf, W_tree, b_tree, W_a1, b_a1, W_a2, b_a2,
        bottom_idx, out);
}